// GConv_87883620811274
// MI455X (gfx1250) — compile-verified
//
#include <hip/hip_runtime.h>

#define N_NODES 50000
#define N_EDGES 800000
#define HID 128
#define BN_EPS 1e-5f

typedef __attribute__((ext_vector_type(2))) float v2f;
typedef __attribute__((ext_vector_type(8))) float v8f;

// ---------------------------------------------------------------- zero
__global__ void k_zero(float* __restrict__ p, int n) {
    int i = blockIdx.x * blockDim.x + threadIdx.x;
    if (i < n) p[i] = 0.f;
}

// ------------------------------------------------- edge scatter-add (agg)
// One edge per 32-lane group; each lane moves 4 consecutive floats.
__global__ __launch_bounds__(256) void k_scatter(const float* __restrict__ z,
                                                 const int* __restrict__ src,
                                                 const int* __restrict__ dst,
                                                 float* __restrict__ agg) {
    int e = blockIdx.x * 8 + (threadIdx.x >> 5);
    if (e >= N_EDGES) return;
    int lane = threadIdx.x & 31;
    int s = src[e], d = dst[e];
    const float4* zr = (const float4*)(z + (size_t)s * HID);
    float4 v = zr[lane];
    float* a = agg + (size_t)d * HID + lane * 4;
    atomicAdd(a + 0, v.x);
    atomicAdd(a + 1, v.y);
    atomicAdd(a + 2, v.z);
    atomicAdd(a + 3, v.w);
}

// ------------------------------------------------- fused WMMA f32 GEMM
// out(N_NODES x 128) = act( (A [+ AGG]) @ W(128x128) + bias )
// Block: 256 threads = 8 waves; wave w computes rows [blk*128 + w*16, +16),
// all 128 output columns as 8 accumulator tiles of V_WMMA_F32_16X16X4_F32.
// LDS weight layout is K-pair interleaved: sW2[(k>>1)*256 + n*2 + (k&1)],
// so each B fragment {W[k][n], W[k+1][n]} is one aligned ds_load_b64 that
// lands directly in an even-aligned VGPR pair (no shuffle movs).
template <bool HAS_AGG, bool RELU>
__global__ __launch_bounds__(256) void k_gemm(const float* __restrict__ A,
                                              const float* __restrict__ AGG,
                                              const float* __restrict__ W,
                                              const float* __restrict__ bias,
                                              float* __restrict__ out) {
    __shared__ float sW2[HID * HID];  // 64 KB, K-pair interleaved
    {
        // thread i handles (pair p = i>>7, col n = i&127): coalesced global
        // reads of rows 2p and 2p+1, one ds_store_b64 per element pair.
        for (int i = threadIdx.x; i < (HID / 2) * HID; i += 256) {
            int p = i >> 7;
            int n = i & (HID - 1);
            float2 w;
            w.x = W[(2 * p) * HID + n];
            w.y = W[(2 * p + 1) * HID + n];
            *(float2*)&sW2[p * 2 * HID + n * 2] = w;
        }
    }
    __syncthreads();

    const int wave = threadIdx.x >> 5;
    const int lane = threadIdx.x & 31;
    const int half = lane >> 4;   // 0: lanes 0-15 carry K=0,1 ; 1: lanes 16-31 carry K=2,3
    const int l16  = lane & 15;
    const int row0 = blockIdx.x * 128 + wave * 16;

    // A fragment addressing: lane carries A[row0+l16][k + half*2 + {0,1}]
    int arow = row0 + l16;
    if (arow > N_NODES - 1) arow = N_NODES - 1;  // clamp, keep EXEC all-ones
    const float* Ar = A + (size_t)arow * HID + half * 2;
    const float* Gr = AGG + (size_t)arow * HID + half * 2;

    // Loop-invariant per-lane LDS base; all (k, tile) offsets are immediates.
    const float* Wb = sW2 + half * 2 * HID + l16 * 2;

    v8f acc[8] = {};

    #pragma unroll 4
    for (int k = 0; k < HID; k += 4) {
        float2 av = *(const float2*)(Ar + k);
        if (HAS_AGG) {
            float2 gv = *(const float2*)(Gr + k);
            av.x += gv.x;
            av.y += gv.y;
        }
        v2f a;
        a[0] = av.x;
        a[1] = av.y;
        const float* Wk = Wb + (k >> 1) * 2 * HID;
        #pragma unroll
        for (int t = 0; t < 8; ++t) {
            v2f b = *(const v2f*)(Wk + t * 32);  // {W[k+half*2][n], W[k+half*2+1][n]}
            acc[t] = __builtin_amdgcn_wmma_f32_16x16x4_f32(
                false, a, false, b, (short)0, acc[t], false, false);
        }
    }

    // Epilogue: C/D layout -> VGPR r, lanes 0-15: M=r, lanes 16-31: M=r+8; N=l16+t*16
    #pragma unroll
    for (int t = 0; t < 8; ++t) {
        int n = t * 16 + l16;
        float bv = bias[n];
        #pragma unroll
        for (int r = 0; r < 8; ++r) {
            int row = row0 + r + half * 8;
            if (row < N_NODES) {
                float v = acc[t][r] + bv;
                if (RELU) v = fmaxf(v, 0.f);
                out[(size_t)row * HID + n] = v;
            }
        }
    }
}

// ------------------------------------------------- column stats (sum, sumsq)
__global__ __launch_bounds__(128) void k_colstats(const float* __restrict__ z,
                                                  float* __restrict__ stats) {
    int c = threadIdx.x;  // one column per thread, coalesced row reads
    int chunk = (N_NODES + gridDim.x - 1) / gridDim.x;
    int r0 = blockIdx.x * chunk;
    int r1 = r0 + chunk;
    if (r1 > N_NODES) r1 = N_NODES;
    float s = 0.f, ss = 0.f;
    for (int r = r0; r < r1; ++r) {
        float v = z[(size_t)r * HID + c];
        s += v;
        ss += v * v;
    }
    atomicAdd(&stats[c], s);
    atomicAdd(&stats[HID + c], ss);
}

// ------------------------------------------------- BN apply (+ optional PReLU)
template <bool PRELU>
__global__ void k_bn_apply(const float* __restrict__ in, float* __restrict__ out,
                           const float* __restrict__ stats,
                           const float* __restrict__ gamma,
                           const float* __restrict__ beta,
                           const float* __restrict__ prelu_a) {
    int idx = blockIdx.x * blockDim.x + threadIdx.x;
    if (idx >= N_NODES * HID) return;
    int c = idx & (HID - 1);
    const float inv_n = 1.0f / (float)N_NODES;
    float mean = stats[c] * inv_n;
    float var  = stats[HID + c] * inv_n - mean * mean;  // biased variance
    float v = (in[idx] - mean) * rsqrtf(var + BN_EPS) * gamma[c] + beta[c];
    if (PRELU) {
        float a = prelu_a[0];
        v = (v > 0.f) ? v : a * v;
    }
    out[idx] = v;
}

// ---------------------------------------------------------------- launch
extern "C" void kernel_launch(void* const* d_in, const int* in_sizes, int n_in,
                              void* d_out, int out_size, void* d_ws, size_t ws_size,
                              hipStream_t stream) {
    const float* x   = (const float*)d_in[0];
    const int*   ei  = (const int*)d_in[1];
    const int*   src = ei;
    const int*   dst = ei + N_EDGES;
    const float* w0a = (const float*)d_in[2];
    const float* b0a = (const float*)d_in[3];
    const float* w0b = (const float*)d_in[4];
    const float* b0b = (const float*)d_in[5];
    const float* w1a = (const float*)d_in[6];
    const float* b1a = (const float*)d_in[7];
    const float* w1b = (const float*)d_in[8];
    const float* b1b = (const float*)d_in[9];
    const float* bng = (const float*)d_in[10];
    const float* bnb = (const float*)d_in[11];
    const float* pw  = (const float*)d_in[12];
    const float* pb  = (const float*)d_in[13];
    const float* pbg = (const float*)d_in[14];
    const float* pbb = (const float*)d_in[15];
    const float* pra = (const float*)d_in[16];

    float* out_z = (float*)d_out;
    float* out_p = out_z + (size_t)N_NODES * HID;

    float* agg   = (float*)d_ws;                       // 25.6 MB
    float* tmp   = agg + (size_t)N_NODES * HID;        // 25.6 MB
    float* stats = tmp + (size_t)N_NODES * HID;        // 256 floats

    const int nelem = N_NODES * HID;
    const int zgrid = (nelem + 255) / 256;
    const int sgrid = (N_EDGES + 7) / 8;
    const int ggrid = (N_NODES + 127) / 128;

    // ---- GIN layer 0: z1 = relu( relu((x+agg)W0a+b0a) W0b + b0b )
    k_zero<<<zgrid, 256, 0, stream>>>(agg, nelem);
    k_scatter<<<sgrid, 256, 0, stream>>>(x, src, dst, agg);
    k_gemm<true, true><<<ggrid, 256, 0, stream>>>(x, agg, w0a, b0a, tmp);
    k_gemm<false, true><<<ggrid, 256, 0, stream>>>(tmp, nullptr, w0b, b0b, out_p); // z1 staged in p slot

    // ---- GIN layer 1: z2
    k_zero<<<zgrid, 256, 0, stream>>>(agg, nelem);
    k_scatter<<<sgrid, 256, 0, stream>>>(out_p, src, dst, agg);
    k_gemm<true, true><<<ggrid, 256, 0, stream>>>(out_p, agg, w1a, b1a, tmp);
    k_gemm<false, true><<<ggrid, 256, 0, stream>>>(tmp, nullptr, w1b, b1b, out_z); // z2

    // ---- encoder BatchNorm -> z (in place)
    k_zero<<<1, 256, 0, stream>>>(stats, 2 * HID);
    k_colstats<<<512, 128, 0, stream>>>(out_z, stats);
    k_bn_apply<false><<<zgrid, 256, 0, stream>>>(out_z, out_z, stats, bng, bnb, nullptr);

    // ---- projection head: p = prelu(BN(z @ pw + pb))
    k_gemm<false, false><<<ggrid, 256, 0, stream>>>(out_z, nullptr, pw, pb, out_p);
    k_zero<<<1, 256, 0, stream>>>(stats, 2 * HID);
    k_colstats<<<512, 128, 0, stream>>>(out_p, stats);
    k_bn_apply<true><<<zgrid, 256, 0, stream>>>(out_p, out_p, stats, pbg, pbb, pra);
}